// CustomGAT_62268435857659
// MI455X (gfx1250) — compile-verified
//
#include <hip/hip_runtime.h>
#include <math.h>

typedef __attribute__((ext_vector_type(2))) float v2f;
typedef __attribute__((ext_vector_type(8))) float v8f;

#define DFEAT 64
#define NEG_SLOPE_F 0.2f

__device__ __forceinline__ float lrelu(float x) { return x > 0.f ? x : NEG_SLOPE_F * x; }

// Portable float atomic-max via sign-split int/uint ordering trick.
__device__ __forceinline__ void atomic_max_float(float* addr, float v) {
  if (v >= 0.f) {
    atomicMax((int*)addr, __float_as_int(v));
  } else {
    atomicMin((unsigned int*)addr, __float_as_uint(v));
  }
}

// ---------------------------------------------------------------------------
// Kernel 1: h = act(X) @ W, plus fused alpha_s = h.a_src, alpha_d = h.a_dst.
// One wave computes a 16-row x 64-col strip with fp32 WMMA
// (V_WMMA_F32_16X16X4_F32, K stepped by 4, 4 column tiles -> 64 WMMA/wave).
// All 16 A fragments are preloaded into registers so global loads pipeline;
// W (64x64 fp32, 16KB) is staged in LDS, shared by the 8 waves of a block.
// RELU is a template parameter so the layer-0 path carries no select code.
// ---------------------------------------------------------------------------
template<int RELU>
__global__ __launch_bounds__(256) void gat_gemm_wmma(
    const float* __restrict__ X, const float* __restrict__ W,
    const float* __restrict__ asrc, const float* __restrict__ adst,
    float* __restrict__ H, float* __restrict__ alpha_s, float* __restrict__ alpha_d,
    int N)
{
  __shared__ float lW[DFEAT * DFEAT];
  const int tid = threadIdx.x;
  for (int i = tid; i < DFEAT * DFEAT; i += 256) lW[i] = W[i];
  __syncthreads();

  const int wave = tid >> 5;
  const int lane = tid & 31;
  const int rowBase = (blockIdx.x * 8 + wave) * 16;
  if (rowBase >= N) return;               // wave-uniform: EXEC stays all-ones

  const int half = lane >> 4;             // 0: lanes 0-15, 1: lanes 16-31
  const int l16  = lane & 15;
  const int kOfs = half * 2;              // A/B VGPR0 holds K+0 (lo half) / K+2 (hi half)
  const int aRow = rowBase + l16;         // A fragment: lane -> row M
  const float* __restrict__ xrow = X + (size_t)aRow * DFEAT;

  // Preload all A fragments (pipelined global_load_b64, no per-step stalls).
  v2f af[16];
#pragma unroll
  for (int i = 0; i < 16; ++i)
    af[i] = *reinterpret_cast<const v2f*>(xrow + 4 * i + kOfs);
  if (RELU) {
#pragma unroll
    for (int i = 0; i < 16; ++i) {
      af[i].x = fmaxf(af[i].x, 0.f);
      af[i].y = fmaxf(af[i].y, 0.f);
    }
  }

  v8f acc0 = {}, acc1 = {}, acc2 = {}, acc3 = {};

#pragma unroll
  for (int i = 0; i < 16; ++i) {
    const int k0 = 4 * i;
    const float* wr0 = &lW[(k0 + kOfs) * DFEAT];
    const float* wr1 = wr0 + DFEAT;
    v2f b0, b1, b2, b3;
    b0.x = wr0[l16];      b0.y = wr1[l16];
    b1.x = wr0[16 + l16]; b1.y = wr1[16 + l16];
    b2.x = wr0[32 + l16]; b2.y = wr1[32 + l16];
    b3.x = wr0[48 + l16]; b3.y = wr1[48 + l16];

    acc0 = __builtin_amdgcn_wmma_f32_16x16x4_f32(false, af[i], false, b0, (short)0, acc0, false, false);
    acc1 = __builtin_amdgcn_wmma_f32_16x16x4_f32(false, af[i], false, b1, (short)0, acc1, false, false);
    acc2 = __builtin_amdgcn_wmma_f32_16x16x4_f32(false, af[i], false, b2, (short)0, acc2, false, false);
    acc3 = __builtin_amdgcn_wmma_f32_16x16x4_f32(false, af[i], false, b3, (short)0, acc3, false, false);
  }

  // Epilogue: store 16x64 tile; fused alpha dot-products with 16-lane reductions.
  // C layout: VGPR r, lanes 0-15 -> (M=r, N=lane); lanes 16-31 -> (M=r+8, N=lane-16).
#pragma unroll
  for (int r = 0; r < 8; ++r) {
    const int row = rowBase + r + half * 8;
    const float c0 = acc0[r], c1 = acc1[r], c2 = acc2[r], c3 = acc3[r];
    float* __restrict__ hrow = H + (size_t)row * DFEAT;
    hrow[l16]      = c0;
    hrow[16 + l16] = c1;
    hrow[32 + l16] = c2;
    hrow[48 + l16] = c3;

    float ps = c0 * asrc[l16] + c1 * asrc[16 + l16] + c2 * asrc[32 + l16] + c3 * asrc[48 + l16];
    float pd = c0 * adst[l16] + c1 * adst[16 + l16] + c2 * adst[32 + l16] + c3 * adst[48 + l16];
#pragma unroll
    for (int off = 8; off > 0; off >>= 1) {
      ps += __shfl_xor(ps, off, 16);
      pd += __shfl_xor(pd, off, 16);
    }
    if (l16 == 0) { alpha_s[row] = ps; alpha_d[row] = pd; }
  }
}

// --------------------------- segment softmax passes ------------------------
__global__ void node_max_init(const float* __restrict__ as_, const float* __restrict__ ad_,
                              float* __restrict__ m_, int N) {
  int i = blockIdx.x * blockDim.x + threadIdx.x;
  if (i >= N) return;
  m_[i] = lrelu(as_[i] + ad_[i]);         // self-loop contribution seeds the max
}

__global__ void edge_max(const int* __restrict__ src, const int* __restrict__ dst,
                         const float* __restrict__ as_, const float* __restrict__ ad_,
                         float* __restrict__ m_, int E) {
  int i = blockIdx.x * blockDim.x + threadIdx.x;
  if (i >= E) return;
  int s = src[i], d = dst[i];
  atomic_max_float(&m_[d], lrelu(as_[s] + ad_[d]));
}

// den[i] starts from the self-loop term; also cache the self exp weight.
__global__ void node_denom_init(const float* __restrict__ as_, const float* __restrict__ ad_,
                                const float* __restrict__ m_, float* __restrict__ den_,
                                float* __restrict__ selfw_, int N) {
  int i = blockIdx.x * blockDim.x + threadIdx.x;
  if (i >= N) return;
  float we = expf(lrelu(as_[i] + ad_[i]) - m_[i]);
  selfw_[i] = we;
  den_[i] = we;
}

// Accumulate denom; cache per-edge exp weight so the aggregation pass needs
// no transcendentals or alpha gathers.
__global__ void edge_denom(const int* __restrict__ src, const int* __restrict__ dst,
                           const float* __restrict__ as_, const float* __restrict__ ad_,
                           const float* __restrict__ m_, float* __restrict__ den_,
                           float* __restrict__ wexp_, int E) {
  int i = blockIdx.x * blockDim.x + threadIdx.x;
  if (i >= E) return;
  int s = src[i], d = dst[i];
  float we = expf(lrelu(as_[s] + ad_[d]) - m_[d]);
  wexp_[i] = we;
  atomicAdd(&den_[d], we);
}

// out[i,:] = bias + alpha_self * h[i,:]   (initializes the scatter target)
__global__ void node_agg_init(const float* __restrict__ hA,
                              const float* __restrict__ selfw_, const float* __restrict__ den_,
                              const float* __restrict__ bias, float* __restrict__ hB, int N) {
  int gid = blockIdx.x * blockDim.x + threadIdx.x;
  if (gid >= N * DFEAT) return;
  int n = gid >> 6;
  int c = gid & 63;
  float coef = selfw_[n] / den_[n];
  hB[gid] = bias[c] + coef * hA[gid];
}

// One wave per edge; 2 columns per lane (float2 gather of h[src]).
__global__ __launch_bounds__(256) void edge_agg(
    const int* __restrict__ src, const int* __restrict__ dst,
    const float* __restrict__ hA,
    const float* __restrict__ wexp_, const float* __restrict__ den_,
    float* __restrict__ hB, int E) {
  int e = blockIdx.x * 8 + (threadIdx.x >> 5);
  if (e >= E) return;
  int lane = threadIdx.x & 31;
  int s = src[e], d = dst[e];
  float w = wexp_[e] / den_[d];
  const float2* hv = reinterpret_cast<const float2*>(hA + (size_t)s * DFEAT);
  float2 h2 = hv[lane];
  float* o = hB + (size_t)d * DFEAT + 2 * lane;
  atomicAdd(o,     w * h2.x);
  atomicAdd(o + 1, w * h2.y);
}

// ------------------------------ pooling + head -----------------------------
__global__ void pool_init(float* __restrict__ g_, int n) {
  int i = blockIdx.x * blockDim.x + threadIdx.x;
  if (i < n) g_[i] = -INFINITY;
}

__global__ void pool_max(const float* __restrict__ h, const int* __restrict__ batch,
                         float* __restrict__ g_, int N) {
  int gid = blockIdx.x * blockDim.x + threadIdx.x;
  if (gid >= N * DFEAT) return;
  int n = gid >> 6;
  int c = gid & 63;
  atomic_max_float(&g_[(size_t)batch[n] * DFEAT + c], h[gid]);
}

__global__ void logits_lsm(const float* __restrict__ g_, const float* __restrict__ fcw,
                           const float* __restrict__ fcb, float* __restrict__ out,
                           int G, int C) {
  int i = blockIdx.x * blockDim.x + threadIdx.x;
  if (i >= G) return;
  float lg[16];
  float mx = -INFINITY;
  for (int k = 0; k < C; ++k) {
    float s = fcb[k];
    for (int c = 0; c < DFEAT; ++c) {
      float v = g_[(size_t)i * DFEAT + c];
      v = (fabsf(v) <= 3.0e38f) ? v : 0.f;     // guard -inf from empty segments
      s += v * fcw[k * DFEAT + c];
    }
    lg[k] = s;
    mx = fmaxf(mx, s);
  }
  float se = 0.f;
  for (int k = 0; k < C; ++k) se += expf(lg[k] - mx);
  float lse = mx + logf(se);
  for (int k = 0; k < C; ++k) out[(size_t)i * C + k] = lg[k] - lse;
}

// ---------------------------------------------------------------------------
extern "C" void kernel_launch(void* const* d_in, const int* in_sizes, int n_in,
                              void* d_out, int out_size, void* d_ws, size_t ws_size,
                              hipStream_t stream) {
  const float* x     = (const float*)d_in[0];
  const int*   ei    = (const int*)d_in[1];
  const int*   batch = (const int*)d_in[2];
  const float* Ws    = (const float*)d_in[3];
  const float* a_src = (const float*)d_in[4];
  const float* a_dst = (const float*)d_in[5];
  const float* bs    = (const float*)d_in[6];
  const float* fc_w  = (const float*)d_in[7];
  const float* fc_b  = (const float*)d_in[8];
  float* out = (float*)d_out;

  const int N = in_sizes[2];
  const int E = in_sizes[1] / 2;
  const int C = in_sizes[8];
  const int G = out_size / C;

  const int* src = ei;
  const int* dst = ei + E;

  float* ws = (float*)d_ws;
  float* hA    = ws;                             // [N, 64] transformed features
  float* hB    = hA + (size_t)N * DFEAT;         // [N, 64] aggregation target
  float* as_   = hB + (size_t)N * DFEAT;         // [N]
  float* ad_   = as_ + N;                        // [N]
  float* m_    = ad_ + N;                        // [N] segment max
  float* den_  = m_ + N;                         // [N] softmax denom
  float* selfw_= den_ + N;                       // [N] self-loop exp weight
  float* g_    = selfw_ + N;                     // [G, 64] pooled
  float* wexp_ = g_ + (size_t)G * DFEAT;         // [E] per-edge exp weight

  const int nb  = (N + 255) / 256;
  const int eb  = (E + 255) / 256;
  const int elb = (N * DFEAT + 255) / 256;
  const int ewb = (E + 7) / 8;
  const int strips  = (N + 15) / 16;
  const int gblocks = (strips + 7) / 8;

  const float* hin = x;
  for (int l = 0; l < 3; ++l) {
    const float* W   = Ws + (size_t)l * DFEAT * DFEAT;
    const float* asv = a_src + (size_t)l * DFEAT;
    const float* adv = a_dst + (size_t)l * DFEAT;
    const float* bv  = bs + (size_t)l * DFEAT;

    if (l == 0)
      gat_gemm_wmma<0><<<gblocks, 256, 0, stream>>>(hin, W, asv, adv, hA, as_, ad_, N);
    else
      gat_gemm_wmma<1><<<gblocks, 256, 0, stream>>>(hin, W, asv, adv, hA, as_, ad_, N);
    node_max_init<<<nb, 256, 0, stream>>>(as_, ad_, m_, N);
    edge_max<<<eb, 256, 0, stream>>>(src, dst, as_, ad_, m_, E);
    node_denom_init<<<nb, 256, 0, stream>>>(as_, ad_, m_, den_, selfw_, N);
    edge_denom<<<eb, 256, 0, stream>>>(src, dst, as_, ad_, m_, den_, wexp_, E);
    node_agg_init<<<elb, 256, 0, stream>>>(hA, selfw_, den_, bv, hB, N);
    edge_agg<<<ewb, 256, 0, stream>>>(src, dst, hA, wexp_, den_, hB, E);
    hin = hB;
  }

  pool_init<<<(G * DFEAT + 255) / 256, 256, 0, stream>>>(g_, G * DFEAT);
  pool_max<<<elb, 256, 0, stream>>>(hB, batch, g_, N);
  logits_lsm<<<(G + 255) / 256, 256, 0, stream>>>(g_, fc_w, fc_b, out, G, C);
}